// GraphAttentionLayer_1116691497133
// MI455X (gfx1250) — compile-verified
//
#include <hip/hip_runtime.h>

#define B_    8
#define N_    2048
#define FIN_  256
#define FOUT_ 64
#define ALPHA_ 0.2f
#define NEGV  (-2147483648.0f)

typedef float v2f __attribute__((ext_vector_type(2)));
typedef float v4f __attribute__((ext_vector_type(4)));
typedef float v8f __attribute__((ext_vector_type(8)));

static __device__ __forceinline__ v8f wmma_f32(v2f a, v2f b, v8f c) {
  // V_WMMA_F32_16X16X4_F32: D = A(16x4) * B(4x16) + C(16x16), f32 throughout.
  return __builtin_amdgcn_wmma_f32_16x16x4_f32(
      /*neg_a=*/false, a, /*neg_b=*/false, b,
      /*c_mod=*/(short)0, c, /*reuse_a=*/false, /*reuse_b=*/false);
}

// ---------------------------------------------------------------------------
// Kernel 1: Wh = h @ W (per-batch 2048x256 @ 256x64), f32 WMMA 16x16x4.
// Also writes WhT[b][c][row] (transposed copy) so the att@Wh kernel can load
// its B-operand pairs (Wh[j][c], Wh[j+1][c]) as one contiguous b64.
// Block = 128 threads = 4 waves; wave w owns output cols [16w,16w+16).
// A-tile (16x4 f32): lanes 0-15 -> M=lane, VGPR0/1 -> K=0/1;
//                    lanes 16-31 -> M=lane-16, VGPR0/1 -> K=2/3.
// B-tile (4x16 f32): VGPR r: lanes0-15 K=r, lanes16-31 K=r+2, N=lane&15.
// ---------------------------------------------------------------------------
__global__ __launch_bounds__(128) void k_wh(const float* __restrict__ h,
                                            const float* __restrict__ W,
                                            float* __restrict__ Wh,
                                            float* __restrict__ WhT) {
  const int lane = threadIdx.x & 31;
  const int wave = threadIdx.x >> 5;          // 0..3 -> col tile
  const int half = lane >> 4;                 // 0/1
  const int l16  = lane & 15;
  const int rt   = blockIdx.x;                // 0..1023
  const int b    = rt >> 7;                   // 128 row-tiles per batch
  const int m0   = (rt & 127) << 4;
  const int col  = (wave << 4) + l16;

  const float* Arow = h + (size_t)(b * N_ + m0 + l16) * FIN_ + (half << 1);
  const float* Bcol = W + (size_t)(half << 1) * FOUT_ + col;

  v8f acc = {};
#pragma unroll 4
  for (int k = 0; k < FIN_; k += 4) {
    v2f a = *(const v2f*)(Arow + k);                    // K = k+2h, k+2h+1
    v2f bb;
    bb.x = Bcol[(size_t)k * FOUT_];                     // W[k+2h][col]
    bb.y = Bcol[(size_t)(k + 1) * FOUT_];               // W[k+2h+1][col]
    acc = wmma_f32(a, bb, acc);
  }

  // C layout: VGPR r -> rows m0+r (lanes0-15) / m0+r+8 (lanes16-31), N=l16.
  const int rbase = m0 + (half << 3);
  float* Crow = Wh + (size_t)(b * N_ + rbase) * FOUT_ + col;
  float* Tcol = WhT + ((size_t)b * FOUT_ + col) * N_ + rbase;
#pragma unroll
  for (int r = 0; r < 8; ++r) {
    Crow[(size_t)r * FOUT_] = acc[r];   // row-major copy (for k_att_vec)
    Tcol[r] = acc[r];                   // transposed copy (for k_attmm B-op)
  }
}

// ---------------------------------------------------------------------------
// Kernel 2: Wh1 = Wh @ a[:64], Wh2 = Wh @ a[64:]. One thread per row.
// ---------------------------------------------------------------------------
__global__ __launch_bounds__(256) void k_att_vec(const float* __restrict__ Wh,
                                                 const float* __restrict__ a,
                                                 float* __restrict__ Wh1,
                                                 float* __restrict__ Wh2) {
  const int rid = blockIdx.x * 256 + threadIdx.x;       // 0..B*N-1
  const float* row = Wh + (size_t)rid * FOUT_;
  float s1 = 0.f, s2 = 0.f;
#pragma unroll
  for (int c = 0; c < FOUT_; c += 4) {
    float4 v  = *(const float4*)(row + c);
    float4 a1 = *(const float4*)(a + c);
    float4 a2 = *(const float4*)(a + FOUT_ + c);
    s1 += v.x * a1.x + v.y * a1.y + v.z * a1.z + v.w * a1.w;
    s2 += v.x * a2.x + v.y * a2.y + v.z * a2.z + v.w * a2.w;
  }
  Wh1[rid] = s1;
  Wh2[rid] = s2;
}

// ---------------------------------------------------------------------------
// Kernel 3: per-(b,j) column softmax stats over i (axis=1 softmax!).
// Block = 256 = 64 cols x 4 i-chunks of 512. Online max/sum, LDS merge.
// Emits packed params P[b][j] = (Wh2[j], L[j]) with L = m + ln(s), so the
// consumer computes att = exp(v - L) with a single b128 per j-pair.
// ---------------------------------------------------------------------------
__global__ __launch_bounds__(256) void k_colstats(const int* __restrict__ adj,
                                                  const float* __restrict__ Wh1,
                                                  const float* __restrict__ Wh2,
                                                  float2* __restrict__ P) {
  const int tx = threadIdx.x & 63;
  const int ty = threadIdx.x >> 6;                      // 0..3
  const int b  = blockIdx.y;
  const int j  = blockIdx.x * 64 + tx;

  const float w2 = Wh2[b * N_ + j];
  const int*   colp = adj + (size_t)b * N_ * N_ + j;
  const float* w1p  = Wh1 + b * N_;

  float m = -__builtin_inff();
  float s = 0.f;
  const int i0 = ty * (N_ / 4);
#pragma unroll 4
  for (int i = i0; i < i0 + N_ / 4; ++i) {
    const int av = colp[(size_t)i * N_];
    float e = w1p[i] + w2;
    e = (e >= 0.f) ? e : ALPHA_ * e;
    const float v = (av > 0) ? e : NEGV;
    if (v > m) { s = s * __expf(m - v) + 1.f; m = v; }
    else       { s += __expf(v - m); }
  }

  __shared__ float sm[4][64];
  __shared__ float ss[4][64];
  sm[ty][tx] = m;
  ss[ty][tx] = s;
  __syncthreads();
  if (ty == 0) {
#pragma unroll
    for (int t = 1; t < 4; ++t) {
      const float m2 = sm[t][tx], s2 = ss[t][tx];
      const float mn = fmaxf(m, m2);
      s = s * __expf(m - mn) + s2 * __expf(m2 - mn);
      m = mn;
    }
    // att(i,j) = exp(v - m)/s = exp(v - (m + ln s)); s in [1, 2048] -> finite.
    P[b * N_ + j] = make_float2(w2, m + __logf(s));
  }
}

// ---------------------------------------------------------------------------
// Kernel 4: out = elu( softatt @ Wh ), attention recomputed on the fly.
// Block = 256 = 8 waves; block owns 16 rows x all 64 cols; wave w owns
// K chunk [256w, 256w+256). Per K-step loads: adj b64 + params b128 +
// 4x WhT b64  ->  6 loads per 4 WMMAs. Cross-wave LDS reduce, elu, store.
// ---------------------------------------------------------------------------
__global__ __launch_bounds__(256) void k_attmm(const int* __restrict__ adj,
                                               const float* __restrict__ WhT,
                                               const float* __restrict__ Wh1,
                                               const float2* __restrict__ P,
                                               float* __restrict__ out) {
  const int lane = threadIdx.x & 31;
  const int wave = threadIdx.x >> 5;          // 0..7 -> K chunk
  const int half = lane >> 4;
  const int l16  = lane & 15;
  const int rt   = blockIdx.x;                // 0..1023
  const int b    = rt >> 7;
  const int m0   = (rt & 127) << 4;
  const int i    = m0 + l16;                  // this lane's A row

  const float w1 = Wh1[b * N_ + i];
  const int*    adjrow = adj + (size_t)(b * N_ + i) * N_;
  const float2* Pp     = P + b * N_;
  const float*  T0 = WhT + ((size_t)b * FOUT_ + l16 +  0) * N_;
  const float*  T1 = WhT + ((size_t)b * FOUT_ + l16 + 16) * N_;
  const float*  T2 = WhT + ((size_t)b * FOUT_ + l16 + 32) * N_;
  const float*  T3 = WhT + ((size_t)b * FOUT_ + l16 + 48) * N_;

  v8f acc0 = {}, acc1 = {}, acc2 = {}, acc3 = {};

  const int kbeg = wave * (N_ / 8);
  const int kend = kbeg + (N_ / 8);
  for (int k = kbeg; k < kend; k += 4) {
    const int j0 = k + (half << 1);           // this lane's K pair (even)
    const int2 av = *(const int2*)(adjrow + j0);
    const v4f  pq = *(const v4f*)(Pp + j0);   // {w2[j0], L[j0], w2[j0+1], L[j0+1]}

    float e0 = w1 + pq.x;
    float e1 = w1 + pq.z;
    e0 = (e0 >= 0.f) ? e0 : ALPHA_ * e0;
    e1 = (e1 >= 0.f) ? e1 : ALPHA_ * e1;
    const float v0 = (av.x > 0) ? e0 : NEGV;  // masked entries still get
    const float v1 = (av.y > 0) ? e1 : NEGV;  // exp(NEG-L) (uniform-col case)
    v2f a;
    a.x = __expf(v0 - pq.y);
    a.y = __expf(v1 - pq.w);

    const v2f b0 = *(const v2f*)(T0 + j0);    // {Wh[j0][c], Wh[j0+1][c]}
    const v2f b1 = *(const v2f*)(T1 + j0);
    const v2f b2 = *(const v2f*)(T2 + j0);
    const v2f b3 = *(const v2f*)(T3 + j0);

    acc0 = wmma_f32(a, b0, acc0);
    acc1 = wmma_f32(a, b1, acc1);
    acc2 = wmma_f32(a, b2, acc2);
    acc3 = wmma_f32(a, b3, acc3);
  }

  __shared__ float red[8][16][64];            // 32 KB
#pragma unroll
  for (int r = 0; r < 8; ++r) {
    const int rr = r + (half << 3);
    red[wave][rr][l16 +  0] = acc0[r];
    red[wave][rr][l16 + 16] = acc1[r];
    red[wave][rr][l16 + 32] = acc2[r];
    red[wave][rr][l16 + 48] = acc3[r];
  }
  __syncthreads();

  float* orow = out + (size_t)(b * N_ + m0) * FOUT_;
  for (int q = threadIdx.x; q < 16 * 64; q += 256) {
    const int r = q >> 6, c = q & 63;
    float sum = 0.f;
#pragma unroll
    for (int w = 0; w < 8; ++w) sum += red[w][r][c];
    sum = (sum > 0.f) ? sum : (expm1f(sum));  // elu, alpha=1
    orow[(size_t)r * FOUT_ + c] = sum;
  }
}

// ---------------------------------------------------------------------------
extern "C" void kernel_launch(void* const* d_in, const int* in_sizes, int n_in,
                              void* d_out, int out_size, void* d_ws, size_t ws_size,
                              hipStream_t stream) {
  const float* h   = (const float*)d_in[0];   // (B,N,FIN)
  const int*   adj = (const int*)d_in[1];     // (B,N,N,1)
  const float* W   = (const float*)d_in[2];   // (FIN,FOUT)
  const float* a   = (const float*)d_in[3];   // (2*FOUT,1)
  float* out = (float*)d_out;                 // (B,N,FOUT)

  // Workspace layout (floats): Wh | WhT | Wh1 | Wh2 | P(float2)  (~8.4 MB)
  float*  Wh  = (float*)d_ws;
  float*  WhT = Wh  + (size_t)B_ * N_ * FOUT_;
  float*  Wh1 = WhT + (size_t)B_ * N_ * FOUT_;
  float*  Wh2 = Wh1 + (size_t)B_ * N_;
  float2* Pp  = (float2*)(Wh2 + (size_t)B_ * N_);

  k_wh      <<<dim3(B_ * N_ / 16), dim3(128), 0, stream>>>(h, W, Wh, WhT);
  k_att_vec <<<dim3(B_ * N_ / 256), dim3(256), 0, stream>>>(Wh, a, Wh1, Wh2);
  k_colstats<<<dim3(N_ / 64, B_), dim3(256), 0, stream>>>(adj, Wh1, Wh2, Pp);
  k_attmm   <<<dim3(B_ * N_ / 16), dim3(256), 0, stream>>>(adj, WhT, Wh1, Pp, out);
}